// SlaterDetFlax_16673063043575
// MI455X (gfx1250) — compile-verified
//
#include <hip/hip_runtime.h>
#include <hip/hip_bf16.h>

// Problem dims (fixed by the reference): B=1024 samples, S=1024 sites,
// NF=128 orbitals (Slater matrix is 128x128), F=128 input features.
#define S_SITES 1024
#define NDIM    128      // NF == F == 128
#define LDA     129      // padded LDS row stride for the Slater matrix (floats)
#define LDB     130      // padded LDS column stride for transposed staged B (floats)
#define MREG    (NDIM * LDB)   // matrix LDS region (dwords): max(128*130, 128*129)

typedef __attribute__((ext_vector_type(2))) float v2f;
typedef __attribute__((ext_vector_type(8))) float v8f;
typedef __attribute__((ext_vector_type(4))) unsigned int v4u;
typedef __attribute__((ext_vector_type(4))) int v4i;
typedef __attribute__((ext_vector_type(8))) int v8i;

union TdmG0 { unsigned long long q[2]; v4u v; };
union TdmG1 { unsigned long long q[4]; v8i v; };

// One block per sample. 256 threads = 8 waves (wave32).
// Phase 0: wave 0 issues a TDM DMA (tensor_load_to_lds) staging the 128x128
//          kernel matrix TRANSPOSED (column-major, stride 130) into LDS,
//          overlapped with the occupancy ballot scan.
// Phase 1: wave 0 extracts occupied site indices in ascending order.
// Phase 2: 8 waves compute the 128x128 Slater matrix with v_wmma_f32_16x16x4_f32;
//          A rows gathered from global features (VMEM port), B fragments are
//          single ds_load_b64's from the transposed staging (DS port).
// Phase 3: LU with partial pivoting in LDS -> slogdet -> complex64 output.
__global__ __launch_bounds__(256)
void slater_logdet_fused(const float* __restrict__ config,    // [B, S]
                         const float* __restrict__ features,  // [B, S, F]
                         const float* __restrict__ kern,      // [F, NF]
                         const float* __restrict__ bias,      // [NF]
                         float2* __restrict__ out)            // [B] complex64
{
    extern __shared__ float smem[];
    float* Asl = smem;                       // NDIM * LDA floats (Slater matrix, phase 3)
    float* Bt  = smem;                       // NDIM cols * LDB floats (B^T staging, phase 2 only;
                                             // unioned with Asl -- dead before Asl is written)
    int*   occ = (int*)(smem + MREG);        // 128 occupied-site indices
    float* sf  = (float*)(occ + NDIM);       // 128 floats  (argmax reduction values)
    int*   si  = (int*)(sf + NDIM);          // 128 ints    (argmax reduction indices)
    float* msc = (float*)(si + NDIM);        // 16 floats   (scalars: inv pivot)

    const int b    = blockIdx.x;
    const int t    = threadIdx.x;
    const int lane = t & 31;
    const int w    = t >> 5;        // wave id 0..7 -> tile band rows [16w, 16w+16)
    const int half = lane >> 4;     // 0: K pair {0,1}, 1: K pair {2,3}
    const int l16  = lane & 15;

    // ---- Phase 0+1: TDM transposed stage of kernel matrix + ballot scan ----
    if (t < 32) {
        // D# for a transposing copy: 3-D view of kern with
        //   X (innermost, contiguous in LDS): 1 element
        //   Y: 128 rows,  memory stride 128 (walks down a column)
        //   Z: 128 cols,  memory stride 1
        // => LDS receives column-major B. Padding: +2 dwords per 128 dwords
        //    stored -> LDS column stride 130 dwords (8B-aligned b64 frags).
        TdmG0 g0;
        const unsigned long long lds_addr = (unsigned int)(uintptr_t)Bt;
        const unsigned long long gaddr    = (unsigned long long)(uintptr_t)kern;
        g0.q[0] = 1ull                                   // count = 1 (valid descriptor)
                | (lds_addr << 32);                      // lds_addr [63:32]
        g0.q[1] = (gaddr & 0x01FFFFFFFFFFFFFFull)        // global_addr [120:64] (57b)
                | (2ull << 62);                          // type = 2 ("image") [127:126]
        TdmG1 g1;
        g1.q[0] = (2ull << 16)                           // data_size = 2 -> 4 bytes
                | (1ull << 20)                           // pad_enable
                | (6ull << 22)                           // pad_interval: 128 dwords
                | (1ull << 25)                           // pad_amount: 2 dwords
                | (1ull << 48);                          // tensor_dim0 = 1   [79:48] low
        g1.q[1] = ((unsigned long long)NDIM << 16)       // tensor_dim1 = 128 [111:80]
                | (1ull << 48);                          // tile_dim0 = 1     [127:112]
        g1.q[2] = (unsigned long long)NDIM               // tile_dim1 = 128   [143:128]
                | ((unsigned long long)NDIM << 16)       // tile_dim2 = 128   [159:144]
                | ((unsigned long long)NDIM << 32);      // tensor_dim0_stride = 128 [207:160] low
        g1.q[3] = (1ull << 16);                          // tensor_dim1_stride = 1 [255:208]
        const v4i g2 = {NDIM, 0, 0, 0};                  // tensor_dim2 = 128; rest 0
        const v4i g3 = {0, 0, 0, 0};
        const v8i gz8 = {0, 0, 0, 0, 0, 0, 0, 0};
        // amdgpu-toolchain (clang-23) 6-arg form: (g0, g1, g2, g3, extra, cpol)
        __builtin_amdgcn_tensor_load_to_lds(g0.v, g1.v, g2, g3, gz8, 0);

        // Occupied indices, ascending order (matches top_k on 0/1 rows).
        const float* cfg = config + (size_t)b * S_SITES;
        int base = 0;
        for (int c = 0; c < S_SITES / 32; ++c) {
            const int s = c * 32 + t;
            const float v = cfg[s];
            const unsigned int m = (unsigned int)__ballot(v != 0.0f);
            const int below = __builtin_popcount(m & ((1u << t) - 1u));
            if (v != 0.0f && (base + below) < NDIM) occ[base + below] = s;
            base += __builtin_popcount(m);
        }
        __builtin_amdgcn_s_wait_tensorcnt(0);            // DMA complete before barrier
    }
    __syncthreads();

    // ---- Phase 2: Slater = gather(features) @ kernel + bias, via WMMA f32 ----
    // A fragment (16x4 f32): lanes 0-15 hold M=0..15 K={k,k+1}; lanes 16-31 K={k+2,k+3}.
    // B fragment (4x16 f32): VGPR0 = row (k+2*half), VGPR1 = row (k+2*half+1), N = l16.
    // Transposed staging puts both B values at Bt[col*LDB + kk .. kk+1] -> one b64.
    const int am = w * 16 + l16;  // A row this lane supplies (same for both halves)
    const float* arow = features + ((size_t)b * S_SITES + occ[am]) * NDIM;
    const float* bbase = Bt + (size_t)l16 * LDB + 2 * half;  // + n*16*LDB + k per step

    v8f acc[8] = {};  // 8 output tiles along N for this wave's band

    for (int k = 0; k < NDIM; k += 4) {
        const float2 af = *(const float2*)(arow + k + 2 * half);
        v2f a; a.x = af.x; a.y = af.y;
        float2 bf[8];
#pragma unroll
        for (int n = 0; n < 8; ++n) {
            bf[n] = *(const float2*)(bbase + (size_t)n * 16 * LDB + k);  // ds_load_b64
        }
#pragma unroll
        for (int n = 0; n < 8; ++n) {
            v2f bb; bb.x = bf[n].x; bb.y = bf[n].y;
            acc[n] = __builtin_amdgcn_wmma_f32_16x16x4_f32(
                false, a, false, bb, (short)0, acc[n], false, false);
        }
    }
    __syncthreads();   // all waves done reading staged B before Asl overwrites it

    // Bias + store tiles to LDS. C/D layout: VGPR r -> row (r + 8*half), col = l16.
    {
        const int row0 = w * 16 + 8 * half;
#pragma unroll
        for (int n = 0; n < 8; ++n) {
            const float bv = bias[n * 16 + l16];
            const int col = n * 16 + l16;
#pragma unroll
            for (int r = 0; r < 8; ++r) {
                Asl[(row0 + r) * LDA + col] = acc[n][r] + bv;
            }
        }
    }
    __syncthreads();

    // ---- Phase 3: LU with partial pivoting, slogdet accumulation ----
    int   neg    = 0;     // sign parity (thread 0 only)
    float logabs = 0.0f;  // sum log|u_kk|  (thread 0 only)

    for (int k = 0; k < NDIM; ++k) {
        // argmax |A[i][k]| over i >= k (column access: conflict-free via LDA=129)
        if (t < NDIM) {
            sf[t] = (t >= k) ? fabsf(Asl[t * LDA + k]) : -1.0f;
            si[t] = t;
        }
        __syncthreads();
        for (int off = 64; off > 0; off >>= 1) {
            if (t < off) {
                if (sf[t + off] > sf[t]) { sf[t] = sf[t + off]; si[t] = si[t + off]; }
            }
            __syncthreads();
        }
        const int p = si[0];

        // row swap k <-> p
        if (t < NDIM && p != k) {
            const float tmp = Asl[k * LDA + t];
            Asl[k * LDA + t] = Asl[p * LDA + t];
            Asl[p * LDA + t] = tmp;
        }
        __syncthreads();

        if (t == 0) {
            const float u = Asl[k * LDA + k];
            logabs += logf(fabsf(u));
            if (u < 0.0f) neg ^= 1;
            if (p != k)   neg ^= 1;
            msc[0] = 1.0f / u;
        }
        __syncthreads();
        const float invp = msc[0];

        // rank-1 update: two threads per row (t and t+128) split the columns
        const int row = t & 127;
        float l = 0.0f;
        if (row > k) l = Asl[row * LDA + k] * invp;
        __syncthreads();
        if (row > k) {
            if (t < NDIM) Asl[row * LDA + k] = l;
            const float* prow = Asl + k * LDA;
            float* rrow = Asl + row * LDA;
            for (int j = k + 1 + (t >> 7); j < NDIM; j += 2) {
                rrow[j] -= l * prow[j];
            }
        }
        __syncthreads();
    }

    if (t == 0) {
        out[b] = make_float2(logabs, neg ? 3.14159265358979323846f : 0.0f);
    }
}

extern "C" void kernel_launch(void* const* d_in, const int* in_sizes, int n_in,
                              void* d_out, int out_size, void* d_ws, size_t ws_size,
                              hipStream_t stream) {
    const float* config   = (const float*)d_in[0];  // [B, S]
    const float* features = (const float*)d_in[1];  // [B, S, F]
    const float* kern     = (const float*)d_in[2];  // [F, NF]
    const float* bias     = (const float*)d_in[3];  // [NF]
    float2* out = (float2*)d_out;                   // [B] complex64 (re, im)

    const int B = in_sizes[0] / S_SITES;

    // LDS: matrix region (128*130, unioned B^T staging / 128*129 Slater)
    //      + occ/sf/si (128 each) + scalars (16)
    const size_t shmem = (size_t)(MREG + NDIM + NDIM + NDIM + 16) * sizeof(float);

    slater_logdet_fused<<<B, 256, shmem, stream>>>(config, features, kern, bias, out);
}